// GATForMultipleChoice_42039139893746
// MI455X (gfx1250) — compile-verified
//
#include <hip/hip_runtime.h>
#include <hip/hip_bf16.h>

#define NA 5
#define NN 10000
#define NE 80000
#define ET (NE + NN)        // edges + self loops = 90000
#define DIN 300
#define H1 8
#define C1 64
#define HC (H1 * C1)        // 512

typedef float v2f __attribute__((ext_vector_type(2)));
typedef float v8f __attribute__((ext_vector_type(8)));

__device__ __forceinline__ float lrelu(float v) { return v > 0.f ? v : 0.2f * v; }

// ---------------------------------------------------------------------------
// K1: h[a] = x[a] @ W1 (WMMA f32 16x16x4) with:
//   - W1 column-block (300x64 = 76.8 KB) staged into LDS once per block via
//     gfx1250 async copy (global_load_async_to_lds_b128 + s_wait_asynccnt)
//   - attention scores a_s/a_d fused into the epilogue (column group == head)
// Block = 128 threads = 4 waves = 4 consecutive M-tiles sharing one column
// group. grid = (ceil(625/4)=157, 8 heads, 5 answers).
// ---------------------------------------------------------------------------
__global__ void __launch_bounds__(128)
gemm_h_kernel(const float* __restrict__ x,
              const float* __restrict__ W1,
              const float* __restrict__ att_s_g,
              const float* __restrict__ att_d_g,
              float* __restrict__ h,
              float* __restrict__ a_s,
              float* __restrict__ a_d) {
  __shared__ __align__(16) float lw[DIN * 64];   // [k][64] column block of W1

  const int tid  = threadIdx.x;
  const int a    = blockIdx.z;
  const int head = blockIdx.y;                   // column group == head
  const int n0   = head * 64;

  // ---- stage W1[:, n0:n0+64] into LDS (async DMA path on gfx1250) ----
#if defined(__gfx1250__)
  {
    const unsigned ldsBase = (unsigned)(size_t)lw;   // LDS aperture: low 32 bits = wg-relative offset
    for (int chunk = tid; chunk < (DIN * 64) / 4; chunk += 128) {
      const int k  = chunk >> 4;                  // row of W1
      const int c4 = (chunk & 15) << 2;           // column quad within group
      const unsigned dst = ldsBase + chunk * 16;  // [k][64] layout == flat chunks
      const unsigned long long src =
          (unsigned long long)(size_t)(W1 + (size_t)k * HC + n0 + c4);
      asm volatile("global_load_async_to_lds_b128 %0, %1, off"
                   :: "v"(dst), "v"(src) : "memory");
    }
    asm volatile("s_wait_asynccnt 0x0" ::: "memory");
  }
#else
  for (int chunk = tid; chunk < (DIN * 64) / 4; chunk += 128) {
    const int k  = chunk >> 4;
    const int c4 = (chunk & 15) << 2;
    *(float4*)&lw[chunk * 4] = *(const float4*)(W1 + (size_t)k * HC + n0 + c4);
  }
#endif
  __syncthreads();

  const int lane  = tid & 31;
  const int mTile = blockIdx.x * 4 + (tid >> 5);
  if (mTile >= 625) return;                      // after barrier: safe

  const float* __restrict__ xa = x + (size_t)a * NN * DIN;
  float* __restrict__ ha       = h + (size_t)a * NN * HC;

  const int colL  = lane & 15;
  const int m     = mTile * 16 + colL;
  const int kHalf = (lane >> 4) * 2;             // lanes 0-15 -> K{0,1}, 16-31 -> K{2,3}

  // att weights for this lane's 4 columns (col, col+16, col+32, col+48)
  float as_t[4], ad_t[4];
#pragma unroll
  for (int t = 0; t < 4; ++t) {
    as_t[t] = att_s_g[n0 + colL + 16 * t];
    ad_t[t] = att_d_g[n0 + colL + 16 * t];
  }

  v8f c0 = {}, c1 = {}, c2 = {}, c3 = {};

  for (int k0 = 0; k0 < DIN; k0 += 4) {
    v2f af;
    af.x = xa[m * DIN + k0 + kHalf + 0];
    af.y = xa[m * DIN + k0 + kHalf + 1];

    const float* __restrict__ l0 = &lw[(k0 + kHalf) * 64 + colL];
    const float* __restrict__ l1 = l0 + 64;

    v2f b0, b1, b2, b3;
    b0.x = l0[ 0]; b0.y = l1[ 0];
    b1.x = l0[16]; b1.y = l1[16];
    b2.x = l0[32]; b2.y = l1[32];
    b3.x = l0[48]; b3.y = l1[48];

    c0 = __builtin_amdgcn_wmma_f32_16x16x4_f32(false, af, false, b0, (short)0, c0, false, false);
    c1 = __builtin_amdgcn_wmma_f32_16x16x4_f32(false, af, false, b1, (short)0, c1, false, false);
    c2 = __builtin_amdgcn_wmma_f32_16x16x4_f32(false, af, false, b2, (short)0, c2, false, false);
    c3 = __builtin_amdgcn_wmma_f32_16x16x4_f32(false, af, false, b3, (short)0, c3, false, false);
  }

  // ---- store h tile; C/D layout: VGPR i holds rows i (lanes 0-15) / i+8 ----
  const int rowOff = (lane >> 4) * 8;
#pragma unroll
  for (int i = 0; i < 8; ++i) {
    const int mm = mTile * 16 + i + rowOff;
    float* __restrict__ row = ha + (size_t)mm * HC + n0 + colL;
    row[ 0] = c0[i];
    row[16] = c1[i];
    row[32] = c2[i];
    row[48] = c3[i];
  }

  // ---- fused attention scores: reduce 64 channels across the 16-lane half ----
#pragma unroll
  for (int i = 0; i < 8; ++i) {
    float s = c0[i] * as_t[0] + c1[i] * as_t[1] + c2[i] * as_t[2] + c3[i] * as_t[3];
    float d = c0[i] * ad_t[0] + c1[i] * ad_t[1] + c2[i] * ad_t[2] + c3[i] * ad_t[3];
#pragma unroll
    for (int o = 1; o <= 8; o <<= 1) {           // stays within each 16-lane half
      s += __shfl_xor(s, o, 32);
      d += __shfl_xor(d, o, 32);
    }
    if (colL == 0) {
      const int mm = mTile * 16 + i + rowOff;
      a_s[((size_t)a * NN + mm) * H1 + head] = s;
      a_d[((size_t)a * NN + mm) * H1 + head] = d;
    }
  }
}

// ---------------------------------------------------------------------------
// K3: layer-1 softmax denominators (segment_max dropped: shift-invariant)
// ---------------------------------------------------------------------------
__global__ void edge_den1_kernel(const int* __restrict__ ei,
                                 const float* __restrict__ a_s,
                                 const float* __restrict__ a_d,
                                 float* __restrict__ den1) {
  const int a = blockIdx.y;
  const int t = blockIdx.x * blockDim.x + threadIdx.x;
  if (t >= ET * H1) return;
  const int head = t & 7;
  const int e    = t >> 3;
  int src, dst;
  if (e < NE) { src = ei[a * 2 * NE + e]; dst = ei[a * 2 * NE + NE + e]; }
  else        { src = dst = e - NE; }
  const float v = lrelu(a_s[(a * NN + src) * H1 + head] + a_d[(a * NN + dst) * H1 + head]);
  atomicAdd(&den1[(a * NN + dst) * H1 + head], __expf(v));
}

// ---------------------------------------------------------------------------
// K4: layer-1 aggregation: agg[dst] += alpha * h[src]
// ---------------------------------------------------------------------------
__global__ void edge_agg1_kernel(const int* __restrict__ ei,
                                 const float* __restrict__ a_s,
                                 const float* __restrict__ a_d,
                                 const float* __restrict__ den1,
                                 const float* __restrict__ h,
                                 float* __restrict__ agg) {
  const int a = blockIdx.y;
  const int t = blockIdx.x * blockDim.x + threadIdx.x;
  if (t >= ET * 128) return;            // 128 = 8 heads * 16 quads
  const int q = t & 127;
  const int e = t >> 7;
  const int head = q >> 4;
  const int c4   = q & 15;
  int src, dst;
  if (e < NE) { src = ei[a * 2 * NE + e]; dst = ei[a * 2 * NE + NE + e]; }
  else        { src = dst = e - NE; }
  const float v = lrelu(a_s[(a * NN + src) * H1 + head] + a_d[(a * NN + dst) * H1 + head]);
  const float alpha = __expf(v) / (den1[(a * NN + dst) * H1 + head] + 1e-16f);
  const float4 hv = *(const float4*)(h + (size_t)(a * NN + src) * HC + head * C1 + c4 * 4);
  float* __restrict__ o = agg + (size_t)(a * NN + dst) * HC + head * C1 + c4 * 4;
  atomicAdd(o + 0, alpha * hv.x);
  atomicAdd(o + 1, alpha * hv.y);
  atomicAdd(o + 2, alpha * hv.z);
  atomicAdd(o + 3, alpha * hv.w);
}

// ---------------------------------------------------------------------------
// K5: p[n] = sum_c relu(agg[n,c] + b1[c]) * W2[c]   (layer-2 features, 1x1)
// ---------------------------------------------------------------------------
__global__ void node_p_kernel(const float* __restrict__ agg,
                              const float* __restrict__ b1,
                              const float* __restrict__ W2,
                              float* __restrict__ p) {
  const int wid  = (blockIdx.x * blockDim.x + threadIdx.x) >> 5;
  const int lane = threadIdx.x & 31;
  const float* __restrict__ ar = agg + (size_t)wid * HC;
  float s = 0.f;
#pragma unroll
  for (int i = 0; i < 16; ++i) {
    const int c = lane + i * 32;
    float v = ar[c] + b1[c];
    v = v > 0.f ? v : 0.f;
    s += v * W2[c];
  }
#pragma unroll
  for (int o = 16; o > 0; o >>= 1) s += __shfl_xor(s, o, 32);
  if (lane == 0) p[wid] = s;
}

// ---------------------------------------------------------------------------
// K6: layer-2 softmax denominators (scalar head)
// ---------------------------------------------------------------------------
__global__ void edge_den2_kernel(const int* __restrict__ ei,
                                 const float* __restrict__ p,
                                 const float* __restrict__ as2,
                                 const float* __restrict__ ad2,
                                 float* __restrict__ den2) {
  const int a = blockIdx.y;
  const int e = blockIdx.x * blockDim.x + threadIdx.x;
  if (e >= ET) return;
  int src, dst;
  if (e < NE) { src = ei[a * 2 * NE + e]; dst = ei[a * 2 * NE + NE + e]; }
  else        { src = dst = e - NE; }
  const float v = lrelu(p[a * NN + src] * as2[0] + p[a * NN + dst] * ad2[0]);
  atomicAdd(&den2[a * NN + dst], __expf(v));
}

// ---------------------------------------------------------------------------
// K7: layer-2 aggregation collapsed into the node-mean: block-reduce then one
// atomic per block into logit_sum[a]
// ---------------------------------------------------------------------------
__global__ void edge_out2_kernel(const int* __restrict__ ei,
                                 const float* __restrict__ p,
                                 const float* __restrict__ as2,
                                 const float* __restrict__ ad2,
                                 const float* __restrict__ den2,
                                 float* __restrict__ lsum) {
  __shared__ float red[256];
  const int a = blockIdx.y;
  const int e = blockIdx.x * blockDim.x + threadIdx.x;
  float contrib = 0.f;
  if (e < ET) {
    int src, dst;
    if (e < NE) { src = ei[a * 2 * NE + e]; dst = ei[a * 2 * NE + NE + e]; }
    else        { src = dst = e - NE; }
    const float ps = p[a * NN + src];
    const float v  = lrelu(ps * as2[0] + p[a * NN + dst] * ad2[0]);
    contrib = __expf(v) / (den2[a * NN + dst] + 1e-16f) * ps;
  }
  red[threadIdx.x] = contrib;
  __syncthreads();
#pragma unroll
  for (int s = 128; s > 0; s >>= 1) {
    if ((int)threadIdx.x < s) red[threadIdx.x] += red[threadIdx.x + s];
    __syncthreads();
  }
  if (threadIdx.x == 0) atomicAdd(&lsum[a], red[0]);
}

// ---------------------------------------------------------------------------
// K8: logits + log_softmax
// ---------------------------------------------------------------------------
__global__ void finalize_kernel(const float* __restrict__ lsum,
                                const float* __restrict__ b2,
                                float* __restrict__ out) {
  if (threadIdx.x == 0) {
    float l[NA];
    float mx = -1e30f;
    for (int a = 0; a < NA; ++a) {
      l[a] = lsum[a] / (float)NN + b2[0];
      mx = fmaxf(mx, l[a]);
    }
    float s = 0.f;
    for (int a = 0; a < NA; ++a) s += __expf(l[a] - mx);
    const float lse = mx + __logf(s);
    for (int a = 0; a < NA; ++a) out[a] = l[a] - lse;
  }
}

extern "C" void kernel_launch(void* const* d_in, const int* in_sizes, int n_in,
                              void* d_out, int out_size, void* d_ws, size_t ws_size,
                              hipStream_t stream) {
  const float* x    = (const float*)d_in[0];
  const int*   ei   = (const int*)d_in[1];
  const float* W1   = (const float*)d_in[2];
  const float* as1  = (const float*)d_in[3];
  const float* ad1  = (const float*)d_in[4];
  const float* b1   = (const float*)d_in[5];
  const float* W2   = (const float*)d_in[6];
  const float* as2  = (const float*)d_in[7];
  const float* ad2  = (const float*)d_in[8];
  const float* b2   = (const float*)d_in[9];
  float* out = (float*)d_out;

  float* ws = (float*)d_ws;
  size_t off = 0;
  float* agg  = ws + off; off += (size_t)NA * NN * HC;  // zeroed
  float* den1 = ws + off; off += (size_t)NA * NN * H1;  // zeroed
  float* den2 = ws + off; off += (size_t)NA * NN;       // zeroed
  float* lsum = ws + off; off += 8;                     // zeroed
  const size_t zeroBytes = off * sizeof(float);
  float* h    = ws + off; off += (size_t)NA * NN * HC;
  float* a_s  = ws + off; off += (size_t)NA * NN * H1;
  float* a_d  = ws + off; off += (size_t)NA * NN * H1;
  float* p    = ws + off; off += (size_t)NA * NN;

  hipMemsetAsync(ws, 0, zeroBytes, stream);

  // K1: GEMM + fused attention scores. 157x8x5 blocks of 4 waves.
  gemm_h_kernel<<<dim3(157, 8, NA), 128, 0, stream>>>(x, W1, as1, ad1, h, a_s, a_d);

  // K3: layer-1 denominators. threads/answer = 720000.
  edge_den1_kernel<<<dim3((ET * H1 + 255) / 256, NA), 256, 0, stream>>>(ei, a_s, a_d, den1);

  // K4: layer-1 aggregation. threads/answer = 11,520,000.
  edge_agg1_kernel<<<dim3(ET * 128 / 256, NA), 256, 0, stream>>>(ei, a_s, a_d, den1, h, agg);

  // K5: layer-2 features. waves = 50000.
  node_p_kernel<<<6250, 256, 0, stream>>>(agg, b1, W2, p);

  // K6: layer-2 denominators.
  edge_den2_kernel<<<dim3((ET + 255) / 256, NA), 256, 0, stream>>>(ei, p, as2, ad2, den2);

  // K7: layer-2 aggregation folded into node mean.
  edge_out2_kernel<<<dim3((ET + 255) / 256, NA), 256, 0, stream>>>(ei, p, as2, ad2, den2, lsum);

  // K8: log_softmax of 5 logits.
  finalize_kernel<<<1, 32, 0, stream>>>(lsum, b2, out);
}